// ParameterizedConv_21474836480478
// MI455X (gfx1250) — compile-verified
//
#include <hip/hip_runtime.h>

// ---------------------------------------------------------------------------
// ParameterizedConv on MI455X (gfx1250, wave32).
//
//   Z[n, d, i] = sum_k xyz_aug[n,k,d] * neis[n,k,i]      (d=0..3, xyz_aug[3]=1)
//   out[n, :]  = Z[n, 0:128] @ Waug[128, 32]             (Waug rows 96..127 = b)
//
// Phase 1: one wave per point (lane = channel). Neighbor idx/mask are made
//          wave-uniform via v_readlane -> scalar branch + SMEM coords loads;
//          feats row gather is one coalesced 128B vector load per neighbor.
// Phase 2: waves 0/1 run out[16x32] = Z[16x128] @ Waug[128x32] as 32 chained
//          V_WMMA_F32_16X16X4_F32 ops. B fragments (same 16KB for all blocks,
//          L2-hot) are preloaded into registers BEFORE the barrier so they
//          overlap the latency-bound gather phase.
// Data is f32 and the problem is L2-gather bound (~40MB total, all resident
// in the 192MB L2), so f32 WMMA is the right precision: exactness for free.
// ---------------------------------------------------------------------------

typedef __attribute__((ext_vector_type(2))) float v2f;
typedef __attribute__((ext_vector_type(8))) float v8f;

#define NPTS 20000
#define KNBR 13
#define INP 32
#define OUTC 32
#define PTS_PER_BLK 16
#define KAUG 128          // 4 * INP augmented contraction length
#define ZPITCH 132        // 128 + 4 pad -> conflict-free strided LDS reads

__global__ __launch_bounds__(512)
void pconv_wmma_kernel(const float* __restrict__ feats,
                       const float* __restrict__ coords,
                       const float* __restrict__ W,
                       const float* __restrict__ bias,
                       const int* __restrict__ nbr_idx,
                       const unsigned char* __restrict__ nbr_mask,
                       float* __restrict__ out)
{
    __shared__ float ldsZ[PTS_PER_BLK * ZPITCH];

    const int lane = threadIdx.x & 31;
    const int wave = threadIdx.x >> 5;
    const int blockBase = blockIdx.x * PTS_PER_BLK;   // 20000 = 1250 * 16 exactly

    const int hi  = lane >> 4;            // lane half select (WMMA layouts)
    const int m   = lane & 15;
    const int colBase = wave * 16;        // GEMM wave 0: cols 0-15, wave 1: 16-31
    const int col = colBase + m;

    // ---- Preload B fragments (Waug rows r = d*32+i) into registers.
    //      Independent of phase 1 -> issues during the gather.           ----
    v2f bfrag[KAUG / 4];
    if (wave < 2) {
        #pragma unroll
        for (int kk = 0; kk < KAUG / 4; ++kk) {
            const int d  = kk >> 3;           // augmented dim, uniform per chunk
            const int i0 = (kk & 7) * 4;      // input-channel base within d
            const float* wbase = (d < 3) ? (W + d * (INP * OUTC)) : bias;
            const int iA = i0 + (hi ? 2 : 0); // mirrored K interleave (see A)
            bfrag[kk].x = wbase[(iA + 0) * OUTC + col];
            bfrag[kk].y = wbase[(iA + 1) * OUTC + col];
        }
    }

    // ---------------- Phase 1: build Z for this wave's point -----------------
    {
        const int n = blockBase + wave;
        // lanes 0..12 preload this point's neighbor indices / masks
        int myIdx = 0, myMsk = 0;
        if (lane < KNBR) {
            myIdx = nbr_idx[n * KNBR + lane];
            myMsk = (int)nbr_mask[n * KNBR + lane];   // jax bool = 1 byte
        }
        float z0 = 0.f, z1 = 0.f, z2 = 0.f, z3 = 0.f;
        #pragma unroll
        for (int k = 0; k < KNBR; ++k) {
            // readlane -> SGPR: scalar branch, SMEM-eligible coords address
            const int idxk = __builtin_amdgcn_readlane(myIdx, k);
            const int mk   = __builtin_amdgcn_readlane(myMsk, k);
            if (mk) {
                const float f  = feats[idxk * INP + lane];   // coalesced 128B
                const float x  = coords[idxk * 3 + 0];       // uniform (s_load)
                const float y  = coords[idxk * 3 + 1];
                const float zc = coords[idxk * 3 + 2];
                z0 = fmaf(x,  f, z0);
                z1 = fmaf(y,  f, z1);
                z2 = fmaf(zc, f, z2);
                z3 += f;                                     // bias row
            }
        }
        float* zp = &ldsZ[wave * ZPITCH];
        zp[0 * INP + lane] = z0;
        zp[1 * INP + lane] = z1;
        zp[2 * INP + lane] = z2;
        zp[3 * INP + lane] = z3;
    }
    __syncthreads();

    // ------------- Phase 2: out[16x32] = Z[16x128] @ Waug[128x32] ------------
    if (wave < 2) {
        v8f c = {};
        #pragma unroll
        for (int kk = 0; kk < KAUG / 4; ++kk) {
            const int r0 = kk * 4;            // augmented-K base of this chunk
            // A fragment 16x4 (ISA layout): VGPR0 = K r0+{0|2}, VGPR1 = K r0+{1|3}
            const float* za = &ldsZ[m * ZPITCH + r0 + (hi ? 2 : 0)];
            v2f a;
            a.x = za[0];
            a.y = za[1];
            // 8-arg form: (neg_a, A, neg_b, B, c_mod, C, reuse_a, reuse_b)
            c = __builtin_amdgcn_wmma_f32_16x16x4_f32(false, a, false, bfrag[kk],
                                                      (short)0, c, false, false);
        }
        // C/D 16x16 f32 layout: VGPR v holds row M = v + 8*hi, col N = lane&15
        #pragma unroll
        for (int v = 0; v < 8; ++v) {
            const int row = v + (hi ? 8 : 0);
            out[(blockBase + row) * OUTC + col] = c[v];
        }
    }
}

extern "C" void kernel_launch(void* const* d_in, const int* in_sizes, int n_in,
                              void* d_out, int out_size, void* d_ws, size_t ws_size,
                              hipStream_t stream) {
    const float*         feats    = (const float*)d_in[0];
    const float*         coords   = (const float*)d_in[1];
    const float*         W        = (const float*)d_in[2];
    const float*         bias     = (const float*)d_in[3];
    const int*           nbr_idx  = (const int*)d_in[4];
    const unsigned char* nbr_mask = (const unsigned char*)d_in[5];
    float*               out      = (float*)d_out;

    (void)in_sizes; (void)n_in; (void)out_size; (void)d_ws; (void)ws_size;

    dim3 grid(NPTS / PTS_PER_BLK);   // 1250 blocks
    dim3 block(512);                 // 16 waves (wave32): 16 pts, 2 GEMM waves
    pconv_wmma_kernel<<<grid, block, 0, stream>>>(feats, coords, W, bias,
                                                  nbr_idx, nbr_mask, out);
}